// BigNet_65850438582496
// MI455X (gfx1250) — compile-verified
//
#include <hip/hip_runtime.h>

typedef __attribute__((ext_vector_type(2))) float v2f;
typedef __attribute__((ext_vector_type(8))) float v8f;
typedef __attribute__((ext_vector_type(2))) int   v2i;

#define AS1 __attribute__((address_space(1)))
#define AS3 __attribute__((address_space(3)))

#define NLAYERS 1000
#define DIN     100
#define LSTR    102          // padded LDS row stride (floats): spreads banks, keeps 8B align
#define NTILES  7            // ceil(112 / 16) N tiles
#define WAVES   4
#define BLOCK   (WAVES * 32)
#define MROWS   16           // batch rows per wave
#define MTILE   (WAVES * MROWS)

#if __has_builtin(__builtin_amdgcn_global_load_async_to_lds_b64)
#define HAVE_ASYNC_LDS 1
#else
#define HAVE_ASYNC_LDS 0
#endif

__device__ __forceinline__ void wait_async0() {
#if __has_builtin(__builtin_amdgcn_s_wait_asynccnt)
  __builtin_amdgcn_s_wait_asynccnt(0);
#else
  asm volatile("s_wait_asynccnt 0" ::: "memory");
#endif
}

// Stage one 100x100 fp32 weight matrix (row-major) into LDS with padded row
// stride, in b64 chunks spread across the workgroup.
__device__ __forceinline__ void copy_w(const float* __restrict__ src,
                                       float* __restrict__ dst, int tid) {
#if HAVE_ASYNC_LDS
  // Direct memory->LDS DMA per lane; no VGPR round-trip, tracked on ASYNCcnt.
  // Builtin signature (from clang diagnostic): (AS1 v2i*, AS3 v2i*, imm, imm).
#pragma unroll 4
  for (int i = tid; i < 50 * DIN; i += BLOCK) {
    int n = i / 50;
    int c = (i - n * 50) * 2;
    __builtin_amdgcn_global_load_async_to_lds_b64(
        (AS1 v2i*)(AS1 void*)(src + n * DIN + c),
        (AS3 v2i*)(AS3 void*)(dst + n * LSTR + c),
        /*offset=*/0, /*cpol=*/0);
  }
#else
#pragma unroll 4
  for (int i = tid; i < 50 * DIN; i += BLOCK) {
    int n = i / 50;
    int c = (i - n * 50) * 2;
    *(v2f*)&dst[n * LSTR + c] = *(const v2f*)&src[n * DIN + c];
  }
#endif
}

__global__ __launch_bounds__(BLOCK)
void bignet_kernel(const float* __restrict__ x,
                   const float* __restrict__ W,
                   const float* __restrict__ b,
                   const float* __restrict__ Wf,
                   const float* __restrict__ bf,
                   float* __restrict__ out) {
  // Double-buffered weights (rows padded to 112 so N-tile 6 reads zeros),
  // plus one private 16x100 activation tile per wave.
  __shared__ float Wbuf[2][112 * LSTR];
  __shared__ float hbuf[WAVES][MROWS * LSTR];

  const int tid  = threadIdx.x;
  const int lane = tid & 31;
  const int wave = tid >> 5;
  const int half = lane >> 4;   // 0: lanes 0-15, 1: lanes 16-31
  const int l16  = lane & 15;
  float* hrow = &hbuf[wave][0];

  // ---------------- prologue ----------------
  copy_w(W, Wbuf[0], tid);
  // zero the padded weight rows 100..111 of both buffers once (read by N-tile 6)
  for (int i = tid; i < 12 * LSTR; i += BLOCK) {
    Wbuf[0][DIN * LSTR + i] = 0.f;
    Wbuf[1][DIN * LSTR + i] = 0.f;
  }
  // stage this wave's 16x100 slice of x into its activation tile
  {
    const float* xs = x + (size_t)(blockIdx.x * MTILE + wave * MROWS) * DIN;
    for (int i = lane; i < MROWS * 50; i += 32) {
      int m = i / 50;
      int c = (i - m * 50) * 2;
      *(v2f*)&hrow[m * LSTR + c] = *(const v2f*)&xs[m * DIN + c];
    }
  }
  wait_async0();
  __syncthreads();

  // ---------------- 1000 x (Linear + ReLU) ----------------
  for (int l = 0; l < NLAYERS; ++l) {
    const int cur = l & 1;
    // Prefetch next layer's weights into the other buffer; the async DMA
    // overlaps this layer's WMMA work. Safe: all reads of that buffer finished
    // before the previous layer's barrier.
    if (l + 1 < NLAYERS)
      copy_w(W + (size_t)(l + 1) * (DIN * DIN), Wbuf[cur ^ 1], tid);

    const float* __restrict__ Wc   = Wbuf[cur];
    const float* __restrict__ bias = b + (size_t)l * DIN;

    // C init = bias broadcast: lane holds column n = nt*16 + l16 for 8 rows.
    v8f acc[NTILES];
#pragma unroll
    for (int nt = 0; nt < NTILES; ++nt) {
      int n = nt * 16 + l16;
      float bv = 0.f;
      if (n < DIN) bv = bias[n];
      v8f c = {bv, bv, bv, bv, bv, bv, bv, bv};
      acc[nt] = c;
    }

    // K = 100 = 25 steps of K=4. A frag: {h[m][k0], h[m][k0+1]}, k0 = 4s+2*half.
    // B frag: {W[n][k0], W[n][k0+1]}  (B[k][n] = W[n][k], i.e. h @ W^T).
#pragma unroll
    for (int s = 0; s < DIN / 4; ++s) {
      const int k0 = 4 * s + 2 * half;
      v2f a = *(const v2f*)&hrow[l16 * LSTR + k0];
#pragma unroll
      for (int nt = 0; nt < NTILES; ++nt) {
        v2f bb = *(const v2f*)&Wc[(nt * 16 + l16) * LSTR + k0];
        acc[nt] = __builtin_amdgcn_wmma_f32_16x16x4_f32(
            false, a, false, bb, (short)0, acc[nt], false, false);
      }
    }

    // ReLU + scatter back into this wave's tile in A-readable row-major layout.
    // C/D layout: VGPR v holds row m = v + 8*half, column n = nt*16 + l16.
#pragma unroll
    for (int nt = 0; nt < NTILES; ++nt) {
      int n = nt * 16 + l16;
      if (n < DIN) {
#pragma unroll
        for (int v = 0; v < 8; ++v) {
          int m = v + 8 * half;
          hrow[m * LSTR + n] = fmaxf(acc[nt][v], 0.f);
        }
      }
    }
    // Ensure the async prefetch into the other buffer has landed in LDS
    // before any wave can read it next layer.
    wait_async0();
    __syncthreads();
  }

  // ---------------- final Linear(100 -> 10), no ReLU ----------------
  {
    float bv = 0.f;
    if (l16 < 10) bv = bf[l16];
    v8f acc = {bv, bv, bv, bv, bv, bv, bv, bv};
    const v2f bz = {0.f, 0.f};
#pragma unroll
    for (int s = 0; s < DIN / 4; ++s) {
      const int k0 = 4 * s + 2 * half;
      v2f a  = *(const v2f*)&hrow[l16 * LSTR + k0];
      v2f bb = bz;
      if (l16 < 10) bb = *(const v2f*)&Wf[l16 * DIN + k0];  // Wf is 10x100
      acc = __builtin_amdgcn_wmma_f32_16x16x4_f32(
          false, a, false, bb, (short)0, acc, false, false);
    }
    const int row0 = blockIdx.x * MTILE + wave * MROWS;
    if (l16 < 10) {
#pragma unroll
      for (int v = 0; v < 8; ++v) {
        int m = v + 8 * half;
        out[(size_t)(row0 + m) * 10 + l16] = acc[v];
      }
    }
  }
}

extern "C" void kernel_launch(void* const* d_in, const int* in_sizes, int n_in,
                              void* d_out, int out_size, void* d_ws, size_t ws_size,
                              hipStream_t stream) {
  (void)n_in; (void)out_size; (void)d_ws; (void)ws_size;
  const float* x  = (const float*)d_in[0];   // (8192, 100)
  const float* W  = (const float*)d_in[1];   // (1000, 100, 100)
  const float* b  = (const float*)d_in[2];   // (1000, 100)
  const float* Wf = (const float*)d_in[3];   // (10, 100)
  const float* bf = (const float*)d_in[4];   // (10,)
  float* out = (float*)d_out;                // (8192, 10)

  const int Btot = in_sizes[0] / DIN;        // 8192
  dim3 grid(Btot / MTILE);                   // 128 workgroups
  dim3 block(BLOCK);                         // 4 waves (wave32)
  bignet_kernel<<<grid, block, 0, stream>>>(x, W, b, Wf, bf, out);
}